// RayMarcher_76948634075541
// MI455X (gfx1250) — compile-verified
//
#include <hip/hip_runtime.h>
#include <stdint.h>

// Problem shape (from reference setup_inputs): B=4, R=16384, S=96.
constexpr int kS    = 96;    // samples per ray
constexpr int kCH   = 8;     // samples per LDS chunk
constexpr int kNCH  = kS / kCH;
constexpr int kBLK  = 256;   // rays per block (8 wave32)
constexpr int kP    = 6;     // 16B pieces per ray per chunk (8 samples * 12B = 96B)
constexpr int kStrB = 112;   // LDS bytes per ray per chunk (96B data + 16B pad)
constexpr int kStrD = kStrB / 4;

typedef float f4 __attribute__((ext_vector_type(4)));  // 16B-aligned vector

__device__ __forceinline__ void async_ld16(unsigned lds_off, const void* g) {
  // gfx1250 async global->LDS copy, tracked on ASYNCcnt (no VGPR round trip).
  asm volatile("global_load_async_to_lds_b128 %0, %1, off"
               :: "v"(lds_off), "v"(g) : "memory");
}

__global__ void init_ws(unsigned* ws) {
  ws[0] = 0x7F800000u;  // +inf bits  (running min of depths, all > 0)
  ws[1] = 0u;           // 0          (running max of positive floats)
}

// depths sorted along S: global min = min over rays of d[0], max = max of d[S-1].
__global__ void __launch_bounds__(kBLK) depth_minmax(const float* __restrict__ depths,
                                                     unsigned* __restrict__ ws) {
  __shared__ unsigned smin[kBLK], smax[kBLK];
  const int tid = threadIdx.x;
  const int ray = blockIdx.x * kBLK + tid;
  const float* d = depths + (size_t)ray * kS;
  smin[tid] = __float_as_uint(d[0]);
  smax[tid] = __float_as_uint(d[kS - 1]);
  __syncthreads();
  for (int off = kBLK / 2; off > 0; off >>= 1) {
    if (tid < off) {
      smin[tid] = min(smin[tid], smin[tid + off]);
      smax[tid] = max(smax[tid], smax[tid + off]);
    }
    __syncthreads();
  }
  if (tid == 0) {
    atomicMin(&ws[0], smin[0]);
    atomicMax(&ws[1], smax[0]);
  }
}

__device__ __forceinline__ void stage_chunk(const char* cbase, int ray0, int c,
                                            float* buf, int tid) {
  // Cooperatively gather 256 rays x 96B (chunk of 8 samples x 3 floats) into LDS.
  unsigned lb = (unsigned)(uintptr_t)buf;  // low 32 bits of generic shared ptr = LDS byte addr
  #pragma unroll
  for (int j = 0; j < kP; ++j) {
    int u  = j * kBLK + tid;     // 0 .. 1535
    int rl = u / kP;             // local ray
    int pc = u % kP;             // 16B piece within ray-chunk
    const char* g = cbase + (size_t)(ray0 + rl) * (kS * 12)
                          + (size_t)c * 12 + (size_t)pc * 16;
    async_ld16(lb + (unsigned)(rl * kStrB + pc * 16), g);
  }
}

__global__ void __launch_bounds__(kBLK) raymarch(
    const float* __restrict__ colors,
    const float* __restrict__ dens,
    const float* __restrict__ depths,
    float* __restrict__ out_rgb,
    float* __restrict__ out_depth,
    float* __restrict__ out_w,
    const unsigned* __restrict__ ws) {
  __shared__ __align__(16) float ldsc[2][kBLK * kStrD];  // 2 x 28 KB double buffer

  const int tid  = threadIdx.x;
  const int ray0 = blockIdx.x * kBLK;
  const int ray  = ray0 + tid;
  const char* cbase = (const char*)colors;

  // Prologue: DMA chunk 0 into buffer 0.
  stage_chunk(cbase, ray0, 0, &ldsc[0][0], tid);

  const float* drow = depths + (size_t)ray * kS;
  const float* rrow = dens   + (size_t)ray * kS;
  float*       wrow = out_w  + (size_t)ray * kS;

  float T = 1.0f, r0 = 0.f, r1 = 0.f, r2 = 0.f, wsum = 0.f, dsum = 0.f;
  float dprev = drow[0];

  for (int k = 0; k < kNCH; ++k) {
    const int c = k * kCH;
    // Chunk k (issued last iteration / prologue) must have landed in LDS.
    asm volatile("s_wait_asynccnt 0" ::: "memory");
    __syncthreads();
    // Single barrier per chunk: passing it proves every wave finished reading
    // the *other* buffer last iteration, so it is safe to overwrite it now.
    if (k + 1 < kNCH)
      stage_chunk(cbase, ray0, c + kCH, &ldsc[(k + 1) & 1][0], tid);

    // Densities for this chunk: two provably-aligned 128b loads.
    f4 x0 = *(const f4*)(rrow + c);
    f4 x1 = *(const f4*)(rrow + c + 4);
    f4 w0, w1;

    const float* cb = &ldsc[k & 1][tid * kStrD];
    #pragma unroll
    for (int i = 0; i < kCH; ++i) {
      int s = c + i;
      float dnext = 0.f, delta;
      if (s < kS - 1) { dnext = drow[s + 1]; delta = dnext - dprev; }
      else            { delta = 1.0e10f; }
      // softplus(density - 1.0), numerically stable (matches jax.nn.softplus)
      float x  = (i < 4 ? x0[i & 3] : x1[i & 3]) - 1.0f;
      float sp = fmaxf(x, 0.0f) + log1pf(expf(-fabsf(x)));
      float alpha = 1.0f - expf(-delta * sp);
      float w = alpha * T;              // T = exclusive cumprod of (1 - a + eps)
      if (i < 4) w0[i & 3] = w; else w1[i & 3] = w;
      r0 = fmaf(w, cb[i * 3 + 0], r0);
      r1 = fmaf(w, cb[i * 3 + 1], r1);
      r2 = fmaf(w, cb[i * 3 + 2], r2);
      dsum = fmaf(w, dprev, dsum);
      wsum += w;
      T *= (1.0f - alpha + 1e-10f);
      dprev = dnext;
    }
    // Weights out: two provably-aligned 128b stores.
    *(f4*)(wrow + c)     = w0;
    *(f4*)(wrow + c + 4) = w1;
  }

  out_rgb[(size_t)ray * 3 + 0] = r0 * 2.0f - 1.0f;
  out_rgb[(size_t)ray * 3 + 1] = r1 * 2.0f - 1.0f;
  out_rgb[(size_t)ray * 3 + 2] = r2 * 2.0f - 1.0f;

  float df = dsum / wsum;
  if (df != df) df = __builtin_inff();              // NaN -> inf (then clipped to max)
  float gmin = __uint_as_float(ws[0]);
  float gmax = __uint_as_float(ws[1]);
  df = fminf(fmaxf(df, gmin), gmax);
  out_depth[ray] = df * 2.0f - 1.0f;
}

extern "C" void kernel_launch(void* const* d_in, const int* in_sizes, int n_in,
                              void* d_out, int out_size, void* d_ws, size_t ws_size,
                              hipStream_t stream) {
  const float* colors = (const float*)d_in[0];   // [B,R,S,3]
  const float* dens   = (const float*)d_in[1];   // [B,R,S,1]
  const float* depths = (const float*)d_in[2];   // [B,R,S,1] sorted along S
  const int n_rays = in_sizes[1] / kS;           // B*R = 65536

  float* out_rgb   = (float*)d_out;              // [B,R,3]
  float* out_depth = out_rgb + (size_t)n_rays * 3;  // [B,R,1]
  float* out_w     = out_depth + n_rays;            // [B,R,S,1]
  unsigned* ws     = (unsigned*)d_ws;

  const int nb = n_rays / kBLK;
  hipLaunchKernelGGL(init_ws, dim3(1), dim3(1), 0, stream, ws);
  hipLaunchKernelGGL(depth_minmax, dim3(nb), dim3(kBLK), 0, stream, depths, ws);
  hipLaunchKernelGGL(raymarch, dim3(nb), dim3(kBLK), 0, stream,
                     colors, dens, depths, out_rgb, out_depth, out_w, ws);
}